// HeteroGNNLayer_90615220011363
// MI455X (gfx1250) — compile-verified
//
#include <hip/hip_runtime.h>

typedef __attribute__((ext_vector_type(2))) float v2f;
typedef __attribute__((ext_vector_type(8))) float v8f;

#define CH 64

// ---------------------------------------------------------------- degree init
__global__ void init_deg_kernel(float* __restrict__ deg, int n3) {
    int i = blockIdx.x * blockDim.x + threadIdx.x;
    if (i < n3) deg[i] = 1.0f;   // self-loop contributes 1 to every node's degree
}

// count in-degree at dst (edge weight 1)
__global__ void count_deg_kernel(const int* __restrict__ edges, float* __restrict__ deg, int E) {
    int e = blockIdx.x * blockDim.x + threadIdx.x;
    if (e < E) unsafeAtomicAdd(&deg[edges[E + e]], 1.0f);  // row1 = dst
}

__global__ void rsqrt_kernel(float* __restrict__ deg, int n3) {
    int i = blockIdx.x * blockDim.x + threadIdx.x;
    if (i < n3) deg[i] = rsqrtf(deg[i]);   // deg >= 1 always (self-loop)
}

// ---------------------------------------------------------------- h = X @ W via WMMA f32 16x16x4
// one wave per 16-row tile; 4 column tiles of 16; K = 64 in 16 steps of 4
__global__ void __launch_bounds__(32)
gemm_wmma_kernel(const float* __restrict__ X, const float* __restrict__ W,
                 float* __restrict__ H) {
    const int lane = threadIdx.x & 31;
    const int r    = lane & 15;     // M (for A) / N (for B) within tile
    const int kh   = lane >> 4;     // which K-pair this half-wave holds
    const long long m0 = (long long)blockIdx.x * 16;

    const float* xrow = X + (m0 + r) * CH;
    v8f acc0 = {}, acc1 = {}, acc2 = {}, acc3 = {};

    for (int ks = 0; ks < 16; ++ks) {
        const int k0 = ks * 4 + 2 * kh;
        // A fragment: lanes 0-15 -> K = ks*4+{0,1}; lanes 16-31 -> K = ks*4+{2,3}
        v2f a;
        a.x = xrow[k0 + 0];
        a.y = xrow[k0 + 1];
        const float* w0 = W + (long long)k0 * CH;   // row k0   of W
        const float* w1 = w0 + CH;                  // row k0+1 of W
        v2f b;
        b.x = w0[0 * 16 + r]; b.y = w1[0 * 16 + r];
        acc0 = __builtin_amdgcn_wmma_f32_16x16x4_f32(false, a, false, b, (short)0, acc0, false, false);
        b.x = w0[1 * 16 + r]; b.y = w1[1 * 16 + r];
        acc1 = __builtin_amdgcn_wmma_f32_16x16x4_f32(false, a, false, b, (short)0, acc1, false, false);
        b.x = w0[2 * 16 + r]; b.y = w1[2 * 16 + r];
        acc2 = __builtin_amdgcn_wmma_f32_16x16x4_f32(false, a, false, b, (short)0, acc2, false, false);
        b.x = w0[3 * 16 + r]; b.y = w1[3 * 16 + r];
        acc3 = __builtin_amdgcn_wmma_f32_16x16x4_f32(false, a, false, b, (short)0, acc3, false, false);
    }

    // C/D layout: VGPR j -> row j (lanes 0-15) and row j+8 (lanes 16-31), col = lane&15
#pragma unroll
    for (int j = 0; j < 8; ++j) {
        float* hrow = H + (m0 + j + 8 * kh) * CH + r;
        hrow[0]  = acc0[j];
        hrow[16] = acc1[j];
        hrow[32] = acc2[j];
        hrow[48] = acc3[j];
    }
}

// scalar tail for rows not covered by full 16-row tiles (unused when N % 16 == 0)
__global__ void gemm_tail_kernel(const float* __restrict__ X, const float* __restrict__ W,
                                 float* __restrict__ H, int rowStart, int rows) {
    long long t = (long long)blockIdx.x * blockDim.x + threadIdx.x;
    if (t >= (long long)rows * CH) return;
    int row = rowStart + (int)(t >> 6);
    int c   = (int)(t & 63);
    float s = 0.0f;
    for (int k = 0; k < CH; ++k) s += X[(long long)row * CH + k] * W[(long long)k * CH + c];
    H[(long long)row * CH + c] = s;
}

// ---------------------------------------------------------------- out init: bias + self-loop term
__global__ void init_out_kernel(const float* __restrict__ h1, const float* __restrict__ h2,
                                const float* __restrict__ h3,
                                const float* __restrict__ dinv1, const float* __restrict__ dinv2,
                                const float* __restrict__ dinv3,
                                const float* __restrict__ b1, const float* __restrict__ b2,
                                const float* __restrict__ b3,
                                float* __restrict__ out, int N) {
    long long t = (long long)blockIdx.x * blockDim.x + threadIdx.x;
    long long total = (long long)N * CH;
    if (t >= total) return;
    int i = (int)(t >> 6);
    int c = (int)(t & 63);
    float d1 = dinv1[i], d2 = dinv2[i], d3 = dinv3[i];
    out[t]         = b1[c] + b2[c] + d1 * d1 * h1[t] + d2 * d2 * h2[t];  // out_star
    out[total + t] = b3[c] + d3 * d3 * h3[t];                            // out_gal
}

// ---------------------------------------------------------------- edge gather + scatter-sum
// 16 threads per edge, each handling 4 channels (float4)
__global__ void scatter_kernel(const float* __restrict__ H, const float* __restrict__ dinv,
                               const int* __restrict__ edges, float* __restrict__ out, int E) {
    long long t = (long long)blockIdx.x * blockDim.x + threadIdx.x;
    if (t >= (long long)E * 16) return;
    int e  = (int)(t >> 4);
    int c4 = (int)(t & 15);
    int s = edges[e];        // row0 = src
    int d = edges[E + e];    // row1 = dst
    float norm = dinv[s] * dinv[d];
    float4 v = *((const float4*)(H + (long long)s * CH) + c4);
    float* o = out + (long long)d * CH + c4 * 4;
    unsafeAtomicAdd(o + 0, v.x * norm);
    unsafeAtomicAdd(o + 1, v.y * norm);
    unsafeAtomicAdd(o + 2, v.z * norm);
    unsafeAtomicAdd(o + 3, v.w * norm);
}

extern "C" void kernel_launch(void* const* d_in, const int* in_sizes, int n_in,
                              void* d_out, int out_size, void* d_ws, size_t ws_size,
                              hipStream_t stream) {
    const float* x_star = (const float*)d_in[0];
    const float* x_gal  = (const float*)d_in[1];
    const int*   e_ssn  = (const int*)d_in[2];
    const int*   e_ssf  = (const int*)d_in[3];
    const int*   e_ggn  = (const int*)d_in[4];
    const float* W_ssn  = (const float*)d_in[5];
    const float* W_ssf  = (const float*)d_in[6];
    const float* W_ggn  = (const float*)d_in[7];
    const float* b_ssn  = (const float*)d_in[8];
    const float* b_ssf  = (const float*)d_in[9];
    const float* b_ggn  = (const float*)d_in[10];

    const int N = in_sizes[0] / CH;
    const int E = in_sizes[2] / 2;

    float* ws    = (float*)d_ws;
    float* h1    = ws;
    float* h2    = h1 + (size_t)N * CH;
    float* h3    = h2 + (size_t)N * CH;
    float* dinv1 = h3 + (size_t)N * CH;
    float* dinv2 = dinv1 + N;
    float* dinv3 = dinv2 + N;
    float* out   = (float*)d_out;

    // 1) degrees (self-loop = 1) -> dinv, all three edge types
    init_deg_kernel<<<(3 * N + 255) / 256, 256, 0, stream>>>(dinv1, 3 * N);
    count_deg_kernel<<<(E + 255) / 256, 256, 0, stream>>>(e_ssn, dinv1, E);
    count_deg_kernel<<<(E + 255) / 256, 256, 0, stream>>>(e_ssf, dinv2, E);
    count_deg_kernel<<<(E + 255) / 256, 256, 0, stream>>>(e_ggn, dinv3, E);
    rsqrt_kernel<<<(3 * N + 255) / 256, 256, 0, stream>>>(dinv1, 3 * N);

    // 2) h = x @ W via f32 WMMA (full-precision), one wave per 16-row tile
    const int tiles = N / 16;
    const int rem   = N - tiles * 16;
    if (tiles > 0) {
        gemm_wmma_kernel<<<tiles, 32, 0, stream>>>(x_star, W_ssn, h1);
        gemm_wmma_kernel<<<tiles, 32, 0, stream>>>(x_star, W_ssf, h2);
        gemm_wmma_kernel<<<tiles, 32, 0, stream>>>(x_gal,  W_ggn, h3);
    }
    if (rem > 0) {
        int tail = (rem * CH + 255) / 256;
        gemm_tail_kernel<<<tail, 256, 0, stream>>>(x_star, W_ssn, h1, tiles * 16, rem);
        gemm_tail_kernel<<<tail, 256, 0, stream>>>(x_star, W_ssf, h2, tiles * 16, rem);
        gemm_tail_kernel<<<tail, 256, 0, stream>>>(x_gal,  W_ggn, h3, tiles * 16, rem);
    }

    // 3) out = bias + self-loop term
    long long totalNC = (long long)N * CH;
    init_out_kernel<<<(unsigned)((totalNC + 255) / 256), 256, 0, stream>>>(
        h1, h2, h3, dinv1, dinv2, dinv3, b_ssn, b_ssf, b_ggn, out, N);

    // 4) edge scatter-sum (dominant traffic; out + h live in 192MB L2)
    long long work = (long long)E * 16;
    unsigned sblocks = (unsigned)((work + 255) / 256);
    scatter_kernel<<<sblocks, 256, 0, stream>>>(h1, dinv1, e_ssn, out, E);
    scatter_kernel<<<sblocks, 256, 0, stream>>>(h2, dinv2, e_ssf, out, E);
    scatter_kernel<<<sblocks, 256, 0, stream>>>(h3, dinv3, e_ggn, out + totalNC, E);
}